// TruePairwiseAttention1D_670014899015
// MI455X (gfx1250) — compile-verified
//
#include <hip/hip_runtime.h>
#include <math.h>

typedef __attribute__((ext_vector_type(2))) float v2f;
typedef __attribute__((ext_vector_type(4))) float v4f;
typedef __attribute__((ext_vector_type(8))) float v8f;
typedef __attribute__((ext_vector_type(4))) unsigned int v4u;
typedef __attribute__((ext_vector_type(4))) int v4i;
typedef __attribute__((ext_vector_type(8))) int v8i;

constexpr int Bsz   = 4;
constexpr int Cdim  = 256;
constexpr int Lseq  = 2048;
constexpr int Hh    = 8;
constexpr int DHd   = 64;
constexpr int INNER = Hh * DHd;   // 512

// D = A(16x4) * B(4x16) + C, full fp32 WMMA
__device__ __forceinline__ v8f wmma_f32(v2f a, v2f b, v8f c) {
    return __builtin_amdgcn_wmma_f32_16x16x4_f32(
        /*neg_a=*/false, a, /*neg_b=*/false, b,
        /*c_mod=*/(short)0, c, /*reuse_a=*/false, /*reuse_b=*/false);
}

// CDNA5 split-counter waits (memory clobber keeps LDS reads below the wait)
__device__ __forceinline__ void lds_fence() {
    asm volatile("s_wait_dscnt 0" ::: "memory");
}
__device__ __forceinline__ void tensor_wait0() {
    asm volatile("s_wait_tensorcnt 0x0" ::: "memory");
}
__device__ __forceinline__ void tensor_wait2() {
    asm volatile("s_wait_tensorcnt 0x2" ::: "memory");
}

// ---------------------------------------------------------------------------
// TDM: load one 16x64 f32 tile (contiguous rows in global) into LDS with a
// 4-dword pad after every 64 dwords -> LDS row stride 68 floats.
//   group0: count=1 | lds_addr | global_addr(57b) | type=2
//   group1: data_size=4B, pad_enable, pad_interval=5(64dw), pad_amount=3(4dw),
//           tensor_dim0=64, tensor_dim1=16, tile_dim0=64, tile_dim1=16,
//           tensor_dim0_stride=64
// ---------------------------------------------------------------------------
__device__ __forceinline__ void tdm_load_tile16x64(unsigned lds_off,
                                                   const void* gptr,
                                                   v8i g1) {
    unsigned long long ga = (unsigned long long)(uintptr_t)gptr;
    v4u g0;
    g0.x = (unsigned)__builtin_amdgcn_readfirstlane(1);  // count=1
    g0.y = (unsigned)__builtin_amdgcn_readfirstlane((int)lds_off);
    g0.z = (unsigned)__builtin_amdgcn_readfirstlane((int)(unsigned)ga);
    g0.w = (unsigned)__builtin_amdgcn_readfirstlane(
               (int)((((unsigned)(ga >> 32)) & 0x01FFFFFFu) | (2u << 30)));
    v4i z4 = {};
#if defined(__clang_major__) && (__clang_major__ >= 23)
    v8i z8 = {};
    __builtin_amdgcn_tensor_load_to_lds(g0, g1, z4, z4, z8, 0);
#else
    __builtin_amdgcn_tensor_load_to_lds(g0, g1, z4, z4, 0);
#endif
}

__device__ __forceinline__ v8i tdm_group1_16x64() {
    v8i g1;
    g1[0] = (int)((2u << 16)    // data_size = 4B
                | (1u << 20)    // pad_enable
                | (5u << 22)    // pad_interval: 64 dwords
                | (3u << 25));  // pad_amount:   4 dwords
    g1[1] = 64 << 16;           // tensor_dim0 = 64  (bits 79:48, low half)
    g1[2] = 16 << 16;           // tensor_dim1 = 16  (bits 111:80, low half)
    g1[3] = 64 << 16;           // tile_dim0   = 64  (bits 127:112)
    g1[4] = 16;                 // tile_dim1   = 16  (bits 143:128)
    g1[5] = 64;                 // tensor_dim0_stride = 64 (bits 207:160)
    g1[6] = 0;
    g1[7] = 0;
    return g1;
}

// ---------------------------------------------------------------------------
// Kernel 1: Q/K/V projection.  xt(B,L,C) @ W(C,512) -> (B,H,L,DH)
// One wave computes one 16x16 output tile; K-loop over C in steps of 4.
// ---------------------------------------------------------------------------
__global__ __launch_bounds__(256) void qkv_proj(
    const float* __restrict__ x,
    const float* __restrict__ Wq, const float* __restrict__ Wk,
    const float* __restrict__ Wv,
    float* __restrict__ q, float* __restrict__ k, float* __restrict__ v)
{
    const int wave = (blockIdx.x * blockDim.x + threadIdx.x) >> 5;
    const int lane = threadIdx.x & 31;

    const int ntn = INNER / 16;              // 32
    const int nti = Lseq / 16;               // 128
    const int per_mat = Bsz * nti * ntn;     // 16384

    const int which = wave / per_mat;        // 0=Q 1=K 2=V
    int rem = wave % per_mat;
    const int b  = rem / (nti * ntn);  rem %= nti * ntn;
    const int it = rem / ntn;
    const int nt = rem % ntn;

    const float* W   = (which == 0) ? Wq : (which == 1) ? Wk : Wv;
    float*       dst = (which == 0) ? q  : (which == 1) ? k  : v;

    const int i0 = it * 16, n0 = nt * 16;
    const int m     = lane & 15;
    const int khalf = (lane >> 4) * 2;       // lanes 0-15 -> K{0,1}; 16-31 -> K{2,3}
    const int rbase = (lane >> 4) * 8;

    const float* xb = x + (size_t)b * Cdim * Lseq;

    v8f acc = {};
    for (int kk = 0; kk < Cdim; kk += 4) {
        v2f a, bf;
        a.x  = xb[(size_t)(kk + khalf    ) * Lseq + i0 + m];   // coalesced in m
        a.y  = xb[(size_t)(kk + khalf + 1) * Lseq + i0 + m];
        bf.x = W [(size_t)(kk + khalf    ) * INNER + n0 + m];  // coalesced in m
        bf.y = W [(size_t)(kk + khalf + 1) * INNER + n0 + m];
        acc = wmma_f32(a, bf, acc);
    }

    // n0 tile lies within one head (16 | 64)
    const int h = n0 / DHd;
    const int d = (n0 % DHd) + m;
    float* dp = dst + ((size_t)b * Hh + h) * Lseq * DHd + d;
    for (int r = 0; r < 8; ++r) {
        const int row = i0 + rbase + r;
        dp[(size_t)row * DHd] = acc[r];
    }
}

// ---------------------------------------------------------------------------
// Kernel 2: flash attention.  One wave per (b, h, 16-row i-tile).
// K/V tiles arrive via double-buffered TENSOR_LOAD_TO_LDS (TDM pads rows to a
// stride of 68 floats -> conflict-free WMMA B-fragment reads). TDM for tile
// j+1 overlaps WMMA+softmax of tile j; s_wait_tensorcnt<=2 retires the older
// pair (per-wave TDM ops complete in order).
// ---------------------------------------------------------------------------
constexpr int KV_STRIDE = 68;                       // floats, 272B rows
constexpr int TILE_F    = 16 * KV_STRIDE;           // 1088 floats per tile
constexpr int WAVE_F    = 4 * TILE_F + 16 * 17;     // K0,V0,K1,V1 + P scratch

__global__ __launch_bounds__(64) void attn(
    const float* __restrict__ q, const float* __restrict__ k,
    const float* __restrict__ v,
    const float* __restrict__ guidance,
    const unsigned char* __restrict__ pad_mask,
    const float* __restrict__ bias_scale_p,
    float* __restrict__ o)
{
    __shared__ float lds[2 * WAVE_F];               // 2 waves, ~37 KB

    const int wave = threadIdx.x >> 5;
    const int lane = threadIdx.x & 31;

    float* base = &lds[wave * WAVE_F];
    float* ktb[2] = { base,              base + 2 * TILE_F };
    float* vtb[2] = { base + TILE_F,     base + 3 * TILE_F };
    float* pt     = base + 4 * TILE_F;

    const unsigned wbyte = (unsigned)(wave * WAVE_F) * 4u;  // lds block at offset 0
    const unsigned kofs[2] = { wbyte,                  wbyte + 2u * TILE_F * 4u };
    const unsigned vofs[2] = { wbyte + TILE_F * 4u,    wbyte + 3u * TILE_F * 4u };

    const int gwave = blockIdx.x * 2 + wave;
    const int nti = Lseq / 16;                      // 128
    const int b  = gwave / (Hh * nti);
    int rem      = gwave % (Hh * nti);
    const int h  = rem / nti;
    const int it = rem % nti;
    const int i0 = it * 16;

    const float bias_scale = bias_scale_p[0];

    const float* qh = q + ((size_t)b * Hh + h) * Lseq * DHd;
    const float* kh = k + ((size_t)b * Hh + h) * Lseq * DHd;
    const float* vh = v + ((size_t)b * Hh + h) * Lseq * DHd;

    const int m     = lane & 15;
    const int khalf = (lane >> 4) * 2;
    const int rbase = (lane >> 4) * 8;

    const v8i g1 = tdm_group1_16x64();

    // issue TDM pair (K tile, V tile) for j-tile jt into buffer buf
    auto issue_tile = [&](int jt, int buf) {
        tdm_load_tile16x64(kofs[buf], kh + (size_t)jt * 16 * DHd, g1);
        tdm_load_tile16x64(vofs[buf], vh + (size_t)jt * 16 * DHd, g1);
    };

    // Q tile (16 x 64) as 16 A-fragments, loaded once per i-tile (L2 hit)
    v2f qa[16];
    for (int c = 0; c < 16; ++c) {
        const int d = 4 * c + khalf;
        qa[c].x = qh[(size_t)(i0 + m) * DHd + d];
        qa[c].y = qh[(size_t)(i0 + m) * DHd + d + 1];
    }

    float mrow[8], lrow[8];
    for (int r = 0; r < 8; ++r) { mrow[r] = -1e30f; lrow[r] = 0.0f; }
    v8f oacc[4];
    for (int t = 0; t < 4; ++t) oacc[t] = (v8f){};

    const int NJT = Lseq / 16;                      // 128
    issue_tile(0, 0);                               // prologue

    for (int jt = 0; jt < NJT; ++jt) {
        const int j0  = jt * 16;
        const int buf = jt & 1;

        if (jt + 1 < NJT) {
            issue_tile(jt + 1, buf ^ 1);            // prefetch next pair
            tensor_wait2();                         // older pair (this buf) done
        } else {
            tensor_wait0();
        }
        const float* kt = ktb[buf];
        const float* vt = vtb[buf];

        // S = Q * K^T   (B fragment: element(dk, n=j) = K[j, dk])
        v8f s = {};
        for (int c = 0; c < 16; ++c) {
            const int d = 4 * c + khalf;
            v2f bf;
            bf.x = kt[m * KV_STRIDE + d];
            bf.y = kt[m * KV_STRIDE + d + 1];
            s = wmma_f32(qa[c], bf, s);
        }

        // scale + guidance bias + padding mask (finite -1e30, keeps exp clean)
        const float jmask = pad_mask[b * Lseq + j0 + m] ? 0.0f : -1e30f;
        float sv[8];
        for (int r = 0; r < 8; ++r) {
            const int i = i0 + rbase + r;
            float val = s[r] * 0.125f +
                        bias_scale * guidance[((size_t)b * Lseq + i) * Lseq + j0 + m];
            sv[r] = val + jmask;
        }

        // online softmax: row reductions across 16-lane halves (wave32)
        float pr[8];
        for (int r = 0; r < 8; ++r) {
            float rmax = sv[r];
            for (int off = 8; off >= 1; off >>= 1)
                rmax = fmaxf(rmax, __shfl_xor(rmax, off, 32));
            const float mnew = fmaxf(mrow[r], rmax);
            const float corr = expf(mrow[r] - mnew);
            const float p    = expf(sv[r] - mnew);
            float psum = p;
            for (int off = 8; off >= 1; off >>= 1)
                psum += __shfl_xor(psum, off, 32);
            lrow[r] = lrow[r] * corr + psum;
            mrow[r] = mnew;
            pr[r]   = p;
            for (int t = 0; t < 4; ++t) oacc[t][r] *= corr;
        }

        // P: D-layout -> LDS -> A-layout
        for (int r = 0; r < 8; ++r)
            pt[(rbase + r) * 17 + m] = pr[r];
        lds_fence();

        // O += P * V   (A: P rows i, K=j; B: element(j, n=d) = V[j, d])
        for (int c = 0; c < 4; ++c) {
            const int jj = 4 * c + khalf;
            v2f af;
            af.x = pt[m * 17 + jj];
            af.y = pt[m * 17 + jj + 1];
            for (int t = 0; t < 4; ++t) {
                v2f bf;
                bf.x = vt[jj * KV_STRIDE + t * 16 + m];
                bf.y = vt[(jj + 1) * KV_STRIDE + t * 16 + m];
                oacc[t] = wmma_f32(af, bf, oacc[t]);
            }
        }
    }

    // normalize, apply i-row padding mask, store O as (B, L, H, DH)
    for (int r = 0; r < 8; ++r) {
        const int i = i0 + rbase + r;
        const float f = (pad_mask[b * Lseq + i] ? 1.0f : 0.0f) / lrow[r];
        for (int t = 0; t < 4; ++t) {
            const int d = t * 16 + m;
            o[(((size_t)b * Lseq + i) * Hh + h) * DHd + d] = oacc[t][r] * f;
        }
    }
}

// ---------------------------------------------------------------------------
// Kernel 3: out = O(B,L,512) @ Wo(512,256) + bo, stored transposed (B,C,L).
// LDS 16x17 transpose makes the L-major stores coalesced.
// ---------------------------------------------------------------------------
__global__ __launch_bounds__(256) void out_proj(
    const float* __restrict__ o, const float* __restrict__ Wo,
    const float* __restrict__ bo, float* __restrict__ out)
{
    __shared__ float tl[8][16 * 17];

    const int wave = threadIdx.x >> 5;
    const int lane = threadIdx.x & 31;
    const int gw = blockIdx.x * 8 + wave;

    const int ntn = Cdim / 16;               // 16
    const int nti = Lseq / 16;               // 128
    const int b  = gw / (nti * ntn);
    int rem      = gw % (nti * ntn);
    const int it = rem / ntn;
    const int nt = rem % ntn;
    const int i0 = it * 16, n0 = nt * 16;

    const int m     = lane & 15;
    const int khalf = (lane >> 4) * 2;
    const int rbase = (lane >> 4) * 8;

    const float* ob = o + (size_t)b * Lseq * INNER;

    v8f acc = {};
    for (int kk = 0; kk < INNER; kk += 4) {
        v2f a, bf;
        a.x  = ob[(size_t)(i0 + m) * INNER + kk + khalf];
        a.y  = ob[(size_t)(i0 + m) * INNER + kk + khalf + 1];
        bf.x = Wo[(size_t)(kk + khalf    ) * Cdim + n0 + m];   // coalesced
        bf.y = Wo[(size_t)(kk + khalf + 1) * Cdim + n0 + m];
        acc = wmma_f32(a, bf, acc);
    }

    const float bias = bo[n0 + m];
    for (int r = 0; r < 8; ++r)
        tl[wave][(rbase + r) * 17 + m] = acc[r] + bias;
    lds_fence();

    for (int t2 = 0; t2 < 8; ++t2) {
        const int n = t2 * 2 + (lane >> 4);
        const int i = lane & 15;
        out[((size_t)b * Cdim + n0 + n) * Lseq + i0 + i] = tl[wave][i * 17 + n];
    }
}

// ---------------------------------------------------------------------------
extern "C" void kernel_launch(void* const* d_in, const int* in_sizes, int n_in,
                              void* d_out, int out_size, void* d_ws, size_t ws_size,
                              hipStream_t stream) {
    const float*         x          = (const float*)d_in[0];
    const float*         guidance   = (const float*)d_in[1];
    const unsigned char* pad_mask   = (const unsigned char*)d_in[2]; // bool (1B)
    const float*         Wq         = (const float*)d_in[3];
    const float*         Wk         = (const float*)d_in[4];
    const float*         Wv         = (const float*)d_in[5];
    const float*         Wo         = (const float*)d_in[6];
    const float*         bo         = (const float*)d_in[7];
    const float*         bias_scale = (const float*)d_in[8];
    float* out = (float*)d_out;

    const size_t per = (size_t)Bsz * Hh * Lseq * DHd;   // 4,194,304 floats
    float* q = (float*)d_ws;
    float* k = q + per;
    float* v = k + per;
    float* o = v + per;          // (B, L, H*DH) layout; total ws = 67.1 MB

    // 3 * B * (L/16) * (INNER/16) = 49152 tile-waves, 8 waves/block
    qkv_proj<<<6144, 256, 0, stream>>>(x, Wq, Wk, Wv, q, k, v);
    // B * H * (L/16) = 4096 waves, 2 waves/block (double-buffered TDM LDS)
    attn<<<2048, 64, 0, stream>>>(q, k, v, guidance, pad_mask, bias_scale, o);
    // B * (L/16) * (C/16) = 8192 tile-waves, 8 waves/block
    out_proj<<<1024, 256, 0, stream>>>(o, Wo, bo, out);
}